// PMWA_34291018891333
// MI455X (gfx1250) — compile-verified
//
#include <hip/hip_runtime.h>
#include <hip/hip_bf16.h>

// Problem constants (fixed by the reference): N=2, C=128, H=W=256, k=7, m=8.
#define NBATCH 2
#define CTOT   128
#define HW     256
#define K      7
#define K2     49
#define M      8
#define PAD    3
#define HB     32          // H/m
#define WB     32          // W/m
#define CPW    32          // channels per workgroup
#define TILE   14          // M + K - 1 (rows needed)
#define LROW   16          // floats loaded per row (w0-4 .. w0+11), 16B aligned
#define RSTR   20          // LDS row stride in floats (TDM pad: 16 data + 4 pad)
#define XOFF   1           // needed columns are positions 1..14 of the 16 loaded
#define CSTR   (TILE*RSTR) // 280 floats per channel slice

#ifndef __has_builtin
#define __has_builtin(x) 0
#endif
#if __has_builtin(__builtin_amdgcn_tensor_load_to_lds)
#define HAVE_TDM 1
#else
#define HAVE_TDM 0
#endif

#if HAVE_TDM
typedef unsigned int u32x4 __attribute__((ext_vector_type(4)));
typedef int          i32x4 __attribute__((ext_vector_type(4)));
typedef int          i32x8 __attribute__((ext_vector_type(8)));

__device__ __forceinline__ void tdm_issue(u32x4 g0, i32x8 g1, i32x4 g2, i32x4 g3) {
#if defined(__clang_major__) && (__clang_major__ >= 23)
    i32x8 g4 = {0, 0, 0, 0, 0, 0, 0, 0};
    __builtin_amdgcn_tensor_load_to_lds(g0, g1, g2, g3, g4, 0);
#else
    __builtin_amdgcn_tensor_load_to_lds(g0, g1, g2, g3, 0);
#endif
}
#endif

__global__ __launch_bounds__(256) void PMWA_fac_kernel(
    const float* __restrict__ feat,     // [N, C, H, W]
    const float* __restrict__ filt,     // [N, C*49, 32, 32]
    float* __restrict__ out)            // [N, C, H, W]
{
    __shared__ float in_s[CPW * CSTR];  // 8960 floats = 35.8 KB
    __shared__ float w_s [CPW * K2];    // 1568 floats =  6.3 KB

    int b = blockIdx.x;
    const int wb = b & 31; b >>= 5;
    const int hb = b & 31; b >>= 5;
    const int cg = b & 3;  b >>= 2;
    const int n  = b;                   // 0..1
    const int tid    = threadIdx.x;
    const int c_base = cg * CPW;
    const int h0 = hb * M;
    const int w0 = wb * M;

    const bool border = (hb == 0) | (hb == HB - 1) | (wb == 0) | (wb == WB - 1);
#if HAVE_TDM
    const bool async_tile = border;     // interior tiles go through the TDM
#else
    const bool async_tile = true;       // no TDM builtin: everything via async
#endif

#if HAVE_TDM
    // ---------------- TDM descriptors (issued once, by wave 0) ----------------
    if (tid < 32) {
        // ---- weights: 2D tile, 1 element per row, 1568 rows, row stride 1024 elems ----
        {
            unsigned long long ga = (unsigned long long)(
                filt + (size_t)n * (CTOT * K2) * (HB * WB)
                     + (size_t)(c_base * K2) * (HB * WB)
                     + (hb * WB + wb));
            unsigned lds = (unsigned)(unsigned long long)&w_s[0];
            u32x4 g0;
            g0[0] = 1u;                               // count=1, user descriptor
            g0[1] = lds;                              // lds_addr
            g0[2] = (unsigned)ga;                     // global_addr[31:0]
            g0[3] = (unsigned)((ga >> 32) & 0x01FFFFFFu) | 0x80000000u; // ga[56:32] | type=2
            i32x8 g1;
            g1[0] = 0x00020000;                       // data_size=4B; no pad; mask=0
            g1[1] = (int)(1024u << 16);               // tensor_dim0 = 1024 (low16 @ [31:16])
            g1[2] = (int)((6272u & 0xFFFFu) << 16);   // td0 hi=0 | tensor_dim1=6272 low16
            g1[3] = (int)(1u << 16);                  // td1 hi=0 | tile_dim0 = 1
            g1[4] = (int)(CPW * K2);                  // tile_dim1 = 1568 | tile_dim2 = 0
            g1[5] = 1024;                             // tensor_dim0_stride low32
            g1[6] = 0;
            g1[7] = 0;
            i32x4 gz = {0, 0, 0, 0};
            tdm_issue(g0, g1, gz, gz);
        }
        // ---- input tile: 3D 16x14x32 tile with LDS pad 16+4 DWORDs per row ----
        if (!border) {
            unsigned long long ga = (unsigned long long)(
                feat + (((size_t)(n * CTOT + c_base)) * HW + (h0 - PAD)) * HW + (w0 - 4));
            unsigned lds = (unsigned)(unsigned long long)&in_s[0];
            u32x4 g0;
            g0[0] = 1u;
            g0[1] = lds;
            g0[2] = (unsigned)ga;
            g0[3] = (unsigned)((ga >> 32) & 0x01FFFFFFu) | 0x80000000u;
            i32x8 g1;
            // data_size=4B (2<<16) | pad_enable (1<<20) | pad_interval=16DW (3<<22)
            // | pad_amount=4DW (3<<25)
            g1[0] = (int)((2u << 16) | (1u << 20) | (3u << 22) | (3u << 25));
            g1[1] = (int)((unsigned)HW << 16);        // tensor_dim0 = 256
            g1[2] = (int)((unsigned)HW << 16);        // td0 hi=0 | tensor_dim1 = 256
            g1[3] = (int)((unsigned)LROW << 16);      // td1 hi=0 | tile_dim0 = 16
            g1[4] = (int)(TILE | (CPW << 16));        // tile_dim1=14 | tile_dim2=32
            g1[5] = HW;                               // tensor_dim0_stride = 256 (row)
            g1[6] = 0;                                // stride0 hi | stride1 low16 (65536 -> 0)
            g1[7] = 1;                                // tensor_dim1_stride[47:16] = 1 (65536)
            i32x4 g2;
            g2[0] = CTOT;                             // tensor_dim2 = 128
            g2[1] = 0;                                // tensor_dim3 = 0
            g2[2] = 0;                                // tensor_dim2_stride (unused, 3D tile)
            g2[3] = 0;                                // tile_dim3 = 0
            i32x4 g3 = {0, 0, 0, 0};
            tdm_issue(g0, g1, g2, g3);
        }
    }
#else
    // ---- fallback: async gather of per-block weights into LDS ----
    {
        const float* fb = filt
            + (size_t)n * (CTOT * K2) * (HB * WB)
            + (size_t)(c_base * K2) * (HB * WB)
            + (hb * WB + wb);
        for (int idx = tid; idx < CPW * K2; idx += 256) {
            unsigned long long gp = (unsigned long long)(fb + (size_t)idx * (HB * WB));
            unsigned lo = (unsigned)(unsigned long long)&w_s[idx];
            asm volatile("global_load_async_to_lds_b32 %0, %1, off"
                         :: "v"(lo), "v"(gp) : "memory");
        }
    }
#endif

    // ---- border blocks: pre-zero the tile so skipped async lanes leave zeros ----
    if (border) {
        for (int i = tid; i < CPW * CSTR; i += 256) in_s[i] = 0.0f;
        __syncthreads();   // zeros complete before async writes can land
    }

    // ---- async per-element tile load (border blocks, or no-TDM fallback) ----
    if (async_tile) {
        const float* xb = feat + ((size_t)(n * CTOT + c_base)) * HW * HW;
        for (int idx = tid; idx < CPW * TILE * TILE; idx += 256) {
            int c  = idx / (TILE * TILE);
            int r2 = idx - c * (TILE * TILE);
            int y  = r2 / TILE;
            int x  = r2 - y * TILE;
            int h  = h0 - PAD + y;
            int w  = w0 - PAD + x;
            unsigned lo = (unsigned)(unsigned long long)
                          &in_s[c * CSTR + y * RSTR + x + XOFF];
            if ((unsigned)h < (unsigned)HW && (unsigned)w < (unsigned)HW) {
                unsigned long long gp =
                    (unsigned long long)(xb + ((size_t)c * HW + h) * HW + w);
                asm volatile("global_load_async_to_lds_b32 %0, %1, off"
                             :: "v"(lo), "v"(gp) : "memory");
            }
        }
    }

    // ---- drain async + tensor engines, then publish LDS to the workgroup ----
#if __has_builtin(__builtin_amdgcn_s_wait_asynccnt)
    __builtin_amdgcn_s_wait_asynccnt(0);
#else
    asm volatile("s_wait_asynccnt 0" ::: "memory");
#endif
#if HAVE_TDM
    if (tid < 32) {
#if __has_builtin(__builtin_amdgcn_s_wait_tensorcnt)
        __builtin_amdgcn_s_wait_tensorcnt(0);
#else
        asm volatile("s_wait_tensorcnt 0" ::: "memory");
#endif
    }
#endif
    __syncthreads();

    // ---- compute: thread = (channel, row); 8 outputs per thread ----
    const int c_l = tid >> 3;   // 0..31
    const int r   = tid & 7;    // 0..7

    float wv[K2];
#pragma unroll
    for (int t = 0; t < K2; ++t) wv[t] = w_s[c_l * K2 + t];

    float acc[M];
#pragma unroll
    for (int p = 0; p < M; ++p) acc[p] = 0.0f;

    const float* ip = &in_s[c_l * CSTR + r * RSTR];
#pragma unroll
    for (int dh = 0; dh < K; ++dh) {
        float rv[TILE];
#pragma unroll
        for (int x = 0; x < TILE; ++x) rv[x] = ip[dh * RSTR + x + XOFF];
#pragma unroll
        for (int dw = 0; dw < K; ++dw) {
            const float w = wv[dh * K + dw];
#pragma unroll
            for (int p = 0; p < M; ++p)
                acc[p] = fmaf(rv[p + dw], w, acc[p]);
        }
    }

    // ---- store 8 consecutive outputs as two b128 ----
    float* op = out + (((size_t)(n * CTOT + c_base + c_l)) * HW + (h0 + r)) * HW + w0;
    float4 o0 = make_float4(acc[0], acc[1], acc[2], acc[3]);
    float4 o1 = make_float4(acc[4], acc[5], acc[6], acc[7]);
    reinterpret_cast<float4*>(op)[0] = o0;
    reinterpret_cast<float4*>(op)[1] = o1;
}

extern "C" void kernel_launch(void* const* d_in, const int* in_sizes, int n_in,
                              void* d_out, int out_size, void* d_ws, size_t ws_size,
                              hipStream_t stream) {
    (void)in_sizes; (void)n_in; (void)out_size; (void)d_ws; (void)ws_size;
    const float* feat = (const float*)d_in[0];
    const float* filt = (const float*)d_in[1];
    float* out = (float*)d_out;
    // grid = N * channel-groups * Hb * Wb = 2 * 4 * 32 * 32 = 8192 workgroups
    dim3 grid(NBATCH * (CTOT / CPW) * HB * WB);
    PMWA_fac_kernel<<<grid, 256, 0, stream>>>(feat, filt, out);
}